// PointNetSetAbstraction_42331197669887
// MI455X (gfx1250) — compile-verified
//
#include <hip/hip_runtime.h>
#include <hip/hip_bf16.h>
#include <float.h>

typedef float v2f __attribute__((ext_vector_type(2)));
typedef float v8f __attribute__((ext_vector_type(8)));

#define BB     16
#define NN     4096
#define DIN    64
#define SPTS   1024
#define KS     32
#define R2C    0.04f      // radius^2
#define CIN0   67
#define CP0    68         // padded input channels for layer 0
#define MTOT   (BB*SPTS*KS)   // 524288 rows

// ---------------------------------------------------------------------------
// 1) Farthest point sampling: one block per batch, xyz + dist live in LDS.
//    Emits new_xyz directly into d_out (first B*SPTS*3 floats).
// ---------------------------------------------------------------------------
__global__ __launch_bounds__(256) void fps_kernel(const float* __restrict__ xyz,
                                                  float* __restrict__ newxyz) {
  __shared__ float px[NN], py[NN], pz[NN], dist[NN];
  __shared__ float wv[8];
  __shared__ int   wi[8];
  __shared__ int   sFar;
  const int b = blockIdx.x, tid = threadIdx.x;
  const float* base = xyz + (size_t)b * NN * 3;
  for (int p = tid; p < NN; p += 256) {
    px[p] = base[p*3+0]; py[p] = base[p*3+1]; pz[p] = base[p*3+2];
    dist[p] = 1e10f;
  }
  if (tid == 0) sFar = 0;
  __syncthreads();
  const int lane = tid & 31, wave = tid >> 5;
  for (int i = 0; i < SPTS; ++i) {
    const int far = sFar;
    const float cx = px[far], cy = py[far], cz = pz[far];
    if (tid == 0) {
      float* o = newxyz + ((size_t)b * SPTS + i) * 3;
      o[0] = cx; o[1] = cy; o[2] = cz;
    }
    float best = -1.0f; int bidx = 0;
#pragma unroll
    for (int j = 0; j < 16; ++j) {
      const int p = tid + 256 * j;
      const float dx = px[p]-cx, dy = py[p]-cy, dz = pz[p]-cz;
      const float nd = fminf(dist[p], dx*dx + dy*dy + dz*dz);
      dist[p] = nd;
      if (nd > best) { best = nd; bidx = p; }   // j ascending => first occurrence kept
    }
#pragma unroll
    for (int off = 16; off > 0; off >>= 1) {
      const float ov = __shfl_down(best, off);
      const int   oi = __shfl_down(bidx, off);
      if (ov > best || (ov == best && oi < bidx)) { best = ov; bidx = oi; }
    }
    if (lane == 0) { wv[wave] = best; wi[wave] = bidx; }
    __syncthreads();
    if (tid == 0) {
      float bv = wv[0]; int bi = wi[0];
      for (int w = 1; w < 8; ++w)
        if (wv[w] > bv || (wv[w] == bv && wi[w] < bi)) { bv = wv[w]; bi = wi[w]; }
      sFar = bi;
    }
    __syncthreads();
  }
}

// ---------------------------------------------------------------------------
// 2) Ball query + grouping: one wave per query. Ballot-compacted candidate
//    list preserves index order (matches jnp stable argsort tie-break).
//    Writes padded feature rows feat[M][68]: cols 0..2 rel-xyz, 3..66 points,
//    67 zero pad.
// ---------------------------------------------------------------------------
#define QPB 8
#define CAP 1024
__global__ __launch_bounds__(256) void group_kernel(const float* __restrict__ xyz,
                                                    const float* __restrict__ pts,
                                                    const float* __restrict__ newxyz,
                                                    float* __restrict__ feat) {
  __shared__ float candD[QPB][CAP];
  __shared__ int   candI[QPB][CAP];
  const int wave = threadIdx.x >> 5, lane = threadIdx.x & 31;
  const int q = blockIdx.x * QPB + wave;
  const int b = q >> 10;
  const float* xb = xyz + (size_t)b * NN * 3;
  const float* nx = newxyz + (size_t)q * 3;
  const float cx = nx[0], cy = nx[1], cz = nx[2];
  float* cd = candD[wave];
  int*   ci = candI[wave];

  int valid = 0;
  float minD = 3e38f; int minI = 0;
  for (int j0 = 0; j0 < NN; j0 += 32) {
    const int j = j0 + lane;
    const float dx = xb[j*3+0]-cx, dy = xb[j*3+1]-cy, dz = xb[j*3+2]-cz;
    const float d = dx*dx + dy*dy + dz*dz;
    const bool inR = d < R2C;
    const unsigned mask = (unsigned)__ballot(inR);
    if (inR) {
      const int off = valid + __popc(mask & ((1u << lane) - 1u));
      if (off < CAP) { cd[off] = d; ci[off] = j; }
    }
    valid += __popc(mask);
    if (d < minD || (d == minD && j < minI)) { minD = d; minI = j; }
  }
#pragma unroll
  for (int off = 16; off > 0; off >>= 1) {       // wave argmin (closest fallback)
    const float od = __shfl_xor(minD, off);
    const int   oi = __shfl_xor(minI, off);
    if (od < minD || (od == minD && oi < minI)) { minD = od; minI = oi; }
  }
  const int m      = valid < CAP ? valid : CAP;
  const int rounds = valid < KS  ? valid : KS;
  int myIdx = minI;                               // valid==0 fallback: argmin
  float prevD = -1.0f; int prevI = -1;
  for (int r = 0; r < rounds; ++r) {
    float bd = 3e38f; int bi = 0x7fffffff;
    for (int t = lane; t < m; t += 32) {
      const float d = cd[t]; const int i = ci[t];
      if ((d > prevD || (d == prevD && i > prevI)) &&
          (d < bd || (d == bd && i < bi))) { bd = d; bi = i; }
    }
#pragma unroll
    for (int off = 16; off > 0; off >>= 1) {
      const float od = __shfl_xor(bd, off);
      const int   oi = __shfl_xor(bi, off);
      if (od < bd || (od == bd && oi < bi)) { bd = od; bi = oi; }
    }
    if (lane == r) myIdx = bi;
    prevD = bd; prevI = bi;
  }
  if (valid > 0 && valid < KS)                    // wrap: k % valid
    myIdx = __shfl(myIdx, lane % valid);

  const float* pb = pts + (size_t)b * NN * DIN;
  for (int k = 0; k < KS; ++k) {
    const int sel = __shfl(myIdx, k);
    float* row = feat + ((size_t)q * KS + k) * CP0;
    const float* ps = pb + (size_t)sel * DIN;
    float v0;
    if      (lane == 0) v0 = xb[sel*3+0] - cx;
    else if (lane == 1) v0 = xb[sel*3+1] - cy;
    else if (lane == 2) v0 = xb[sel*3+2] - cz;
    else                v0 = ps[lane - 3];
    row[lane]      = v0;
    row[lane + 32] = ps[lane + 29];
    if (lane < 4) row[lane + 64] = (lane < 3) ? ps[lane + 61] : 0.0f;
  }
}

// ---------------------------------------------------------------------------
// 3) fp32 WMMA GEMM: Y[m][o] = act(A)[m] . W[o] + bias[o]. FOLD is a
//    compile-time flag: when true, previous layer's BN+ReLU (per-channel
//    scale/shift, staged in LDS) is applied to A on load. Accumulates
//    per-channel sum / sum-of-squares for this layer's BN in the epilogue.
//    8 waves/block, wave = 16-row strip, V_WMMA_F32_16X16X4_F32 over K.
// ---------------------------------------------------------------------------
template <int COUT, int CINPAD, bool FOLD>
__global__ __launch_bounds__(256) void gemm_bn_kernel(
    const float* __restrict__ A, int cinLog,
    const float* __restrict__ W, const float* __restrict__ bias,
    const float* __restrict__ scIn, const float* __restrict__ shIn,
    float* __restrict__ Y, double* __restrict__ gS, double* __restrict__ gQ) {
  __shared__ float Bm[CINPAD][COUT];     // W^T, zero-padded rows
  __shared__ float biasS[COUT];
  __shared__ float scS[CINPAD], shS[CINPAD];
  __shared__ float pS[8][COUT], pQ[8][COUT];
  const int tid = threadIdx.x;
  for (int idx = tid; idx < CINPAD * COUT; idx += 256) {
    const int c = idx / COUT, o = idx - c * COUT;
    Bm[c][o] = (c < cinLog) ? W[o * cinLog + c] : 0.0f;
  }
  for (int o = tid; o < COUT; o += 256) biasS[o] = bias[o];
  if (FOLD) {
    for (int c = tid; c < CINPAD; c += 256) { scS[c] = scIn[c]; shS[c] = shIn[c]; }
  }
  __syncthreads();

  const int wave = tid >> 5, lane = tid & 31;
  const int laneM = lane & 15, half = lane >> 4;
  const int m0 = blockIdx.x * 128 + wave * 16;
  const int kOff = half * 2;
  constexpr int NT = COUT / 16;
  v8f acc[NT];
#pragma unroll
  for (int nt = 0; nt < NT; ++nt) acc[nt] = (v8f){0,0,0,0,0,0,0,0};

  const float* Arow = A + (size_t)(m0 + laneM) * CINPAD;
#pragma unroll 4
  for (int k = 0; k < CINPAD; k += 4) {
    const int c0 = k + kOff;
    float a0 = Arow[c0], a1 = Arow[c0 + 1];
    if constexpr (FOLD) {
      a0 = fmaxf(a0 * scS[c0]     + shS[c0],     0.0f);
      a1 = fmaxf(a1 * scS[c0 + 1] + shS[c0 + 1], 0.0f);
    }
    v2f a; a.x = a0; a.y = a1;
#pragma unroll
    for (int nt = 0; nt < NT; ++nt) {
      v2f bf; bf.x = Bm[c0][nt*16 + laneM]; bf.y = Bm[c0 + 1][nt*16 + laneM];
      acc[nt] = __builtin_amdgcn_wmma_f32_16x16x4_f32(
          false, a, false, bf, (short)0, acc[nt], false, false);
    }
  }
  // Epilogue: +bias, store raw activations, accumulate BN stats.
#pragma unroll
  for (int nt = 0; nt < NT; ++nt) {
    const int col = nt*16 + laneM;
    const float bv = biasS[col];
    float s = 0.f, qq = 0.f;
#pragma unroll
    for (int r = 0; r < 8; ++r) {
      const float v = acc[nt][r] + bv;
      Y[(size_t)(m0 + r + 8*half) * COUT + col] = v;
      s += v; qq += v * v;
    }
    const float s2 = s  + __shfl_down(s, 16);
    const float q2 = qq + __shfl_down(qq, 16);
    if (half == 0) { pS[wave][col] = s2; pQ[wave][col] = q2; }
  }
  __syncthreads();
  if (tid < COUT) {
    float ts = 0.f, tq = 0.f;
    for (int w = 0; w < 8; ++w) { ts += pS[w][tid]; tq += pQ[w][tid]; }
    unsafeAtomicAdd(&gS[tid], (double)ts);
    unsafeAtomicAdd(&gQ[tid], (double)tq);
  }
}

// ---------------------------------------------------------------------------
// 4) BN stats -> per-channel scale/shift
// ---------------------------------------------------------------------------
__global__ void scaleshift_kernel(const double* __restrict__ gS,
                                  const double* __restrict__ gQ,
                                  const float* __restrict__ gamma,
                                  const float* __restrict__ beta,
                                  float* __restrict__ sc, float* __restrict__ sh,
                                  int cout, double invM) {
  const int c = threadIdx.x;
  if (c < cout) {
    const double m = gS[c] * invM;
    const double v = gQ[c] * invM - m * m;
    const float  s = gamma[c] * rsqrtf((float)v + 1e-5f);
    sc[c] = s;
    sh[c] = beta[c] - (float)m * s;
  }
}

__global__ void zero_kernel(double* p, int n) {
  const int i = blockIdx.x * blockDim.x + threadIdx.x;
  if (i < n) p[i] = 0.0;
}

// ---------------------------------------------------------------------------
// 5) Final BN+ReLU fused with max over the 32 samples.
// ---------------------------------------------------------------------------
__global__ __launch_bounds__(128) void finalize_kernel(const float* __restrict__ X2,
                                                       const float* __restrict__ sc,
                                                       const float* __restrict__ sh,
                                                       float* __restrict__ outp) {
  const int q = blockIdx.x, o = threadIdx.x;
  const float s = sc[o], b = sh[o];
  const float* base = X2 + (size_t)q * KS * 128 + o;
  float mx = -3e38f;
#pragma unroll
  for (int k = 0; k < KS; ++k) mx = fmaxf(mx, base[k*128] * s + b);
  outp[(size_t)q * 128 + o] = fmaxf(mx, 0.0f);
}

// ---------------------------------------------------------------------------
extern "C" void kernel_launch(void* const* d_in, const int* in_sizes, int n_in,
                              void* d_out, int out_size, void* d_ws, size_t ws_size,
                              hipStream_t stream) {
  const float* xyz    = (const float*)d_in[0];
  const float* pts    = (const float*)d_in[1];
  const float* w0     = (const float*)d_in[2];
  const float* b0     = (const float*)d_in[3];
  const float* gamma0 = (const float*)d_in[4];
  const float* beta0  = (const float*)d_in[5];
  const float* w1     = (const float*)d_in[6];
  const float* b1     = (const float*)d_in[7];
  const float* gamma1 = (const float*)d_in[8];
  const float* beta1  = (const float*)d_in[9];
  const float* w2     = (const float*)d_in[10];
  const float* b2     = (const float*)d_in[11];
  const float* gamma2 = (const float*)d_in[12];
  const float* beta2  = (const float*)d_in[13];

  float* out    = (float*)d_out;
  float* newxyz = out;                               // B*SPTS*3
  float* newpts = out + (size_t)BB * SPTS * 3;       // B*SPTS*128

  const size_t M = MTOT;
  float*  feat  = (float*)d_ws;                      // M * 68
  float*  X0    = feat + M * CP0;                    // M * 64
  float*  X2    = X0 + M * 64;                       // M * 128
  double* gS    = (double*)(X2 + M * 128);           // 3 * 128
  double* gQ    = gS + 3 * 128;                      // 3 * 128
  float*  scale = (float*)(gQ + 3 * 128);            // 3 * 128
  float*  shift = scale + 3 * 128;                   // 3 * 128
  float*  X1    = feat;                              // alias (feat dead after L0)

  const double invM = 1.0 / (double)M;
  const int gemmBlocks = (int)(M / 128);

  zero_kernel<<<3, 256, 0, stream>>>(gS, 2 * 3 * 128);
  fps_kernel<<<BB, 256, 0, stream>>>(xyz, newxyz);
  group_kernel<<<(BB * SPTS) / QPB, 256, 0, stream>>>(xyz, pts, newxyz, feat);

  // Layer 0: feat(68 pad, 67 logical) -> X0(64), raw A
  gemm_bn_kernel<64, CP0, false><<<gemmBlocks, 256, 0, stream>>>(
      feat, CIN0, w0, b0, nullptr, nullptr, X0, gS + 0, gQ + 0);
  scaleshift_kernel<<<1, 128, 0, stream>>>(gS + 0, gQ + 0, gamma0, beta0,
                                           scale + 0, shift + 0, 64, invM);
  // Layer 1: BN0+ReLU folded into A-load
  gemm_bn_kernel<64, 64, true><<<gemmBlocks, 256, 0, stream>>>(
      X0, 64, w1, b1, scale + 0, shift + 0, X1, gS + 128, gQ + 128);
  scaleshift_kernel<<<1, 128, 0, stream>>>(gS + 128, gQ + 128, gamma1, beta1,
                                           scale + 128, shift + 128, 64, invM);
  // Layer 2: BN1+ReLU folded into A-load
  gemm_bn_kernel<128, 64, true><<<gemmBlocks, 256, 0, stream>>>(
      X1, 64, w2, b2, scale + 128, shift + 128, X2, gS + 256, gQ + 256);
  scaleshift_kernel<<<1, 128, 0, stream>>>(gS + 256, gQ + 256, gamma2, beta2,
                                           scale + 256, shift + 256, 128, invM);
  // BN2+ReLU + max over K fused
  finalize_kernel<<<BB * SPTS, 128, 0, stream>>>(X2, scale + 256, shift + 256, newpts);
}